// ChebNet_73753178407276
// MI455X (gfx1250) — compile-verified
//
#include <hip/hip_runtime.h>
#include <hip/hip_bf16.h>
#include <math.h>

typedef float v2f __attribute__((ext_vector_type(2)));
typedef float v8f __attribute__((ext_vector_type(8)));

// ---------------------------------------------------------------------------
// K1: zero workspace (deg/dinv [N] + Tx1 [N*128] + Tx2acc [N*128])
// ---------------------------------------------------------------------------
__global__ void zero_ws_kernel(float4* __restrict__ p, long long n4) {
    long long i = (long long)blockIdx.x * blockDim.x + threadIdx.x;
    if (i < n4) p[i] = make_float4(0.f, 0.f, 0.f, 0.f);
}

// ---------------------------------------------------------------------------
// K2: deg[src] += 1 for non-self-loop edges
// ---------------------------------------------------------------------------
__global__ void degree_kernel(const int* __restrict__ src,
                              const int* __restrict__ dst,
                              float* __restrict__ deg, int E) {
    int e = blockIdx.x * blockDim.x + threadIdx.x;
    if (e >= E) return;
    int s = src[e], d = dst[e];
    if (s != d) atomicAdd(&deg[s], 1.0f);
}

// ---------------------------------------------------------------------------
// K3: dinv[i] = deg > 0 ? deg^-0.5 : 0   (in place)
// ---------------------------------------------------------------------------
__global__ void dinv_kernel(float* __restrict__ dinv, int n) {
    int i = blockIdx.x * blockDim.x + threadIdx.x;
    if (i >= n) return;
    float v = dinv[i];
    dinv[i] = (v > 0.0f) ? (1.0f / sqrtf(v)) : 0.0f;
}

// ---------------------------------------------------------------------------
// K4: out[dst] += (-dinv[src]*dinv[dst]) * in[src]   (one wave per edge,
//     each lane handles 4 of the 128 features via float4 gather + f32 atomics;
//     gather source and scatter target both fit in the 192MB L2)
// ---------------------------------------------------------------------------
__global__ void propagate_kernel(const int* __restrict__ src,
                                 const int* __restrict__ dst,
                                 const float* __restrict__ dinv,
                                 const float* __restrict__ in,
                                 float* __restrict__ out, int E) {
    int e = blockIdx.x * 8 + (threadIdx.x >> 5);
    if (e >= E) return;
    int lane = threadIdx.x & 31;
    int s = src[e], d = dst[e];
    if (s == d) return;                       // self loops contribute 0
    float w = -dinv[s] * dinv[d];
    const float4 v = *reinterpret_cast<const float4*>(in + (size_t)s * 128 + lane * 4);
    float* o = out + (size_t)d * 128 + lane * 4;
    atomicAdd(o + 0, w * v.x);
    atomicAdd(o + 1, w * v.y);
    atomicAdd(o + 2, w * v.z);
    atomicAdd(o + 3, w * v.w);
}

// ---------------------------------------------------------------------------
// One K=128 GEMM segment, compile-time specialized so the inner loop has no
// segment-select control flow. SEG==2 builds A = 2*Tx2acc - x on the fly
// (lowers to a single v_pk_fma_f32 with neg modifier).
// ---------------------------------------------------------------------------
template <int SEG>
__device__ __forceinline__ void gemm_seg(const float* __restrict__ A0,
                                         const float* __restrict__ Ax,
                                         const float* __restrict__ W,
                                         size_t arow, int l15, int lhi,
                                         v8f acc[8]) {
    #pragma unroll 4
    for (int k0 = 0; k0 < 128; k0 += 4) {
        const int kk = k0 + 2 * lhi;          // A: lanes 0-15 -> K=k0,k0+1; 16-31 -> k0+2,k0+3
        v2f a;
        const float2 t2 = *reinterpret_cast<const float2*>(A0 + arow + kk);
        if (SEG == 2) {
            const float2 x2 = *reinterpret_cast<const float2*>(Ax + arow + kk);
            a[0] = 2.0f * t2.x - x2.x;
            a[1] = 2.0f * t2.y - x2.y;
        } else {
            a[0] = t2.x;
            a[1] = t2.y;
        }
        const float* __restrict__ Wb = W + (size_t)kk * 128 + l15;  // rows kk, kk+1
        #pragma unroll
        for (int t = 0; t < 8; ++t) {         // 8 N-tiles of 16 cols
            v2f b;
            b[0] = Wb[16 * t];
            b[1] = Wb[128 + 16 * t];
            acc[t] = __builtin_amdgcn_wmma_f32_16x16x4_f32(
                false, a, false, b, (short)0, acc[t], false, false);
        }
    }
}

// ---------------------------------------------------------------------------
// K5: fused  out = relu([x|Tx1|2*Tx2acc-x] @ [W0;W1;W2] + bias) @ fcW + fcB,
//     then row-wise log_softmax. One wave per 16-node tile; WMMA f32 16x16x4.
//     N (100000) is a multiple of 16 so every active wave has full EXEC.
// ---------------------------------------------------------------------------
__global__ void __launch_bounds__(128)
fused_head_kernel(const float* __restrict__ x,
                  const float* __restrict__ tx1,
                  const float* __restrict__ tx2acc,
                  const float* __restrict__ W0,
                  const float* __restrict__ W1,
                  const float* __restrict__ W2,
                  const float* __restrict__ bias,
                  const float* __restrict__ fcW,
                  const float* __restrict__ fcB,
                  float* __restrict__ out, int n) {
    __shared__ float hs[4][16][128];          // 32 KB, one 16x128 tile per wave
    const int wave = threadIdx.x >> 5;
    const int lane = threadIdx.x & 31;
    const int row0 = (blockIdx.x * 4 + wave) * 16;
    if (row0 >= n) return;                    // wave-uniform guard
    const int l15 = lane & 15;
    const int lhi = lane >> 4;                // 0: lanes 0-15, 1: lanes 16-31

    v8f acc[8] = {};                          // 16x128 f32 output tile
    const size_t arow = (size_t)(row0 + l15) * 128;

    gemm_seg<0>(x,      nullptr, W0, arow, l15, lhi, acc);
    gemm_seg<1>(tx1,    nullptr, W1, arow, l15, lhi, acc);
    gemm_seg<2>(tx2acc, x,       W2, arow, l15, lhi, acc);

    // bias + relu -> per-wave LDS tile (C layout: vgpr r holds rows r / r+8)
    #pragma unroll
    for (int t = 0; t < 8; ++t) {
        const int col = 16 * t + l15;
        const float bv = bias[col];
        #pragma unroll
        for (int r = 0; r < 8; ++r) {
            float v = acc[t][r] + bv;
            hs[wave][r + 8 * lhi][col] = (v > 0.0f) ? v : 0.0f;
        }
    }

    // FC (128 -> 40) + log_softmax. 2 lanes per row: each lane does 20 classes.
    const int row = l15;
    const int j0 = lhi * 20;
    float logit[20];
    #pragma unroll
    for (int j = 0; j < 20; ++j) logit[j] = fcB[j0 + j];
    for (int c = 0; c < 128; ++c) {
        const float hv = hs[wave][row][c];
        const float* __restrict__ wc = fcW + (size_t)c * 40 + j0;
        #pragma unroll
        for (int j = 0; j < 20; ++j) logit[j] = fmaf(hv, wc[j], logit[j]);
    }
    float m = logit[0];
    #pragma unroll
    for (int j = 1; j < 20; ++j) m = fmaxf(m, logit[j]);
    m = fmaxf(m, __shfl_xor(m, 16, 32));      // cross-half-wave max
    float s = 0.0f;
    #pragma unroll
    for (int j = 0; j < 20; ++j) s += expf(logit[j] - m);
    s += __shfl_xor(s, 16, 32);               // cross-half-wave sum
    const float lse = m + logf(s);
    float* __restrict__ orow = out + (size_t)(row0 + row) * 40 + j0;
    #pragma unroll
    for (int j = 0; j < 20; ++j) orow[j] = logit[j] - lse;
}

// ---------------------------------------------------------------------------
extern "C" void kernel_launch(void* const* d_in, const int* in_sizes, int n_in,
                              void* d_out, int out_size, void* d_ws, size_t ws_size,
                              hipStream_t stream) {
    const float* x    = (const float*)d_in[0];
    const int*   eidx = (const int*)d_in[1];          // edge_index [2,E] flat
    const float* W0   = (const float*)d_in[2];
    const float* W1   = (const float*)d_in[3];
    const float* W2   = (const float*)d_in[4];
    const float* bias = (const float*)d_in[5];
    const float* fcW  = (const float*)d_in[6];
    const float* fcB  = (const float*)d_in[7];
    float* out = (float*)d_out;

    const int n = in_sizes[0] / 128;                  // 100000
    const int E = in_sizes[1] / 2;                    // 1600000
    const int* src = eidx;
    const int* dst = eidx + E;

    // workspace layout: dinv [n] | Tx1 [n*128] | Tx2acc [n*128]
    float* dinv   = (float*)d_ws;
    float* tx1    = dinv + n;
    float* tx2acc = tx1 + (size_t)n * 128;

    // K1: zero deg + both accumulators
    long long n4 = ((long long)n * 257) / 4;
    zero_ws_kernel<<<(unsigned)((n4 + 255) / 256), 256, 0, stream>>>((float4*)d_ws, n4);
    // K2: degree
    degree_kernel<<<(E + 255) / 256, 256, 0, stream>>>(src, dst, dinv, E);
    // K3: deg -> deg^-1/2
    dinv_kernel<<<(n + 255) / 256, 256, 0, stream>>>(dinv, n);
    // K4a: Tx1 = -A_norm x
    propagate_kernel<<<(E + 7) / 8, 256, 0, stream>>>(src, dst, dinv, x, tx1, E);
    // K4b: Tx2acc = -A_norm Tx1
    propagate_kernel<<<(E + 7) / 8, 256, 0, stream>>>(src, dst, dinv, tx1, tx2acc, E);
    // K5: fused GEMMs (WMMA f32) + bias + relu + FC + log_softmax
    fused_head_kernel<<<(n + 63) / 64, 128, 0, stream>>>(
        x, tx1, tx2acc, W0, W1, W2, bias, fcW, fcB, out, n);
}